// HeteroGCN_30133490549164
// MI455X (gfx1250) — compile-verified
//
#include <hip/hip_runtime.h>
#include <hip/hip_bf16.h>

// ---------------------------------------------------------------------------
// HeteroGCN for MI455X (gfx1250, wave32).
// GEMMs via V_WMMA_F32_16X16X4_F32 (fp32-exact). Scatter via native f32
// global atomics (L2-resident accumulators: 25.6MB << 192MB L2).
// Round 2: templated DOUT (shift-based addressing), scalarized edge loads.
// ---------------------------------------------------------------------------

typedef __attribute__((ext_vector_type(2))) float v2f;
typedef __attribute__((ext_vector_type(8))) float v8f;

#define NN   50000
#define EE   400000
#define RR   3
#define DIN  128
#define DHID 128
#define DOUTF 64

__device__ __forceinline__ void atomAddF(float* p, float v) {
    __hip_atomic_fetch_add(p, v, __ATOMIC_RELAXED, __HIP_MEMORY_SCOPE_AGENT);
}

// ---------------- zero a float buffer ----------------
__global__ void k_zero(float* __restrict__ p, int n) {
    int i = blockIdx.x * blockDim.x + threadIdx.x;
    if (i < n) p[i] = 0.0f;
}

// ---------------- degree counting (both endpoints) ----------------
__global__ void k_degree(const int* __restrict__ src, const int* __restrict__ dst,
                         float* __restrict__ nsrc, float* __restrict__ ndst) {
    int i = blockIdx.x * blockDim.x + threadIdx.x;   // over R*E
    if (i >= RR * EE) return;
    int r = i / EE;
    atomAddF(&nsrc[(size_t)r * NN + src[i]], 1.0f);
    atomAddF(&ndst[(size_t)r * NN + dst[i]], 1.0f);
}

// ---------------- deg -> deg^{-1/2} (0 for deg==0) ----------------
__global__ void k_finalize_norm(float* __restrict__ p, int n) {
    int i = blockIdx.x * blockDim.x + threadIdx.x;
    if (i < n) {
        float v = p[i];
        p[i] = (v > 0.0f) ? rsqrtf(v) : 0.0f;
    }
}

// ---------------- WMMA GEMM:  H = (X * norm_src[:,None]) @ W ----------------
// X: [NN, DIN] fp32, W: [DIN, DOUT] fp32, H: [NN, DOUT] fp32.
// One wave (32 lanes) per 16x16 output tile. grid = (NN/16, DOUT/16), block 32.
template <int DOUT>
__global__ void k_gemm_wmma(const float* __restrict__ X, const float* __restrict__ W,
                            const float* __restrict__ nsrc, float* __restrict__ H) {
    const int lane    = threadIdx.x;           // 0..31
    const int rowBase = blockIdx.x * 16;
    const int colBase = blockIdx.y * 16;

    const int m    = lane & 15;                // A-matrix row for this lane
    const int koff = (lane >> 4) * 2;          // K pair: lanes 0-15 -> K+0/1, 16-31 -> K+2/3
    const int col  = colBase + (lane & 15);    // B-matrix column for this lane

    const float scale = nsrc[rowBase + m];
    const float* __restrict__ xrow = X + (size_t)(rowBase + m) * DIN;
    const float* __restrict__ wcol = W + col;  // column base; row index scales by DOUT

    v8f acc = {};
#pragma unroll
    for (int k = 0; k < DIN; k += 4) {
        v2f a, b;
        a.x = xrow[k + koff]     * scale;
        a.y = xrow[k + koff + 1] * scale;
        b.x = wcol[(k + koff)     * DOUT];
        b.y = wcol[(k + koff + 1) * DOUT];
        acc = __builtin_amdgcn_wmma_f32_16x16x4_f32(
            /*neg_a=*/false, a, /*neg_b=*/false, b,
            /*c_mod=*/(short)0, acc, /*reuse_a=*/false, /*reuse_b=*/false);
    }

    // C/D layout: VGPR v -> (M = v + 8*(lane>=16), N = lane&15)
    const int m0 = (lane >> 4) * 8;
    const int n  = lane & 15;
#pragma unroll
    for (int v = 0; v < 8; ++v) {
        H[(size_t)(rowBase + m0 + v) * DOUT + colBase + n] = acc[v];
    }
}

// ---------------- edge scatter: accum[dst] += H[src] * norm_dst[dst] --------
// One wave per edge; lane handles DOUT/32 consecutive floats.
// Edge index is wave-uniform: force scalar (SMEM) loads via readfirstlane.
template <int DOUT>
__global__ void k_scatter(const float* __restrict__ H, const int* __restrict__ src,
                          const int* __restrict__ dst, const float* __restrict__ ndst,
                          float* __restrict__ accum) {
    constexpr int VEC = DOUT / 32;              // 4 (d=128) or 2 (d=64)
    int e = blockIdx.x * (blockDim.x >> 5) + (threadIdx.x >> 5);
    e = __builtin_amdgcn_readfirstlane(e);      // wave-uniform
    if (e >= EE) return;
    const int lane = threadIdx.x & 31;
    const int s = __builtin_amdgcn_readfirstlane(src[e]);
    const int d = __builtin_amdgcn_readfirstlane(dst[e]);
    const float sc = ndst[d];                   // uniform address -> scalar load

    const int k0 = lane * VEC;
    const float* __restrict__ hsrc = H + (size_t)s * DOUT + k0;
    float* __restrict__ base = accum + (size_t)d * DOUT + k0;

    if (VEC == 4) {
        const float4 h4 = *(const float4*)hsrc;
        atomAddF(base + 0, h4.x * sc);
        atomAddF(base + 1, h4.y * sc);
        atomAddF(base + 2, h4.z * sc);
        atomAddF(base + 3, h4.w * sc);
    } else {
        const float2 h2 = *(const float2*)hsrc;
        atomAddF(base + 0, h2.x * sc);
        atomAddF(base + 1, h2.y * sc);
    }
}

// ---------------- bias + L2 row-normalize + leaky_relu (d == 128) -----------
// One wave per node row; lane holds 4 floats; wave32 shfl reduction.
__global__ void k_bias_normact(const float* __restrict__ accum, const float* __restrict__ b,
                               float* __restrict__ act) {
    const int wave = threadIdx.x >> 5;
    const int lane = threadIdx.x & 31;
    const int node = blockIdx.x * (blockDim.x >> 5) + wave;
    if (node >= NN) return;
    const int k0 = lane * 4;

    float4 v = *(const float4*)(accum + (size_t)node * DHID + k0);
    // bias = sum over R relations
    const float4 b0 = *(const float4*)(b + 0 * DHID + k0);
    const float4 b1 = *(const float4*)(b + 1 * DHID + k0);
    const float4 b2 = *(const float4*)(b + 2 * DHID + k0);
    v.x += b0.x + b1.x + b2.x;
    v.y += b0.y + b1.y + b2.y;
    v.z += b0.z + b1.z + b2.z;
    v.w += b0.w + b1.w + b2.w;

    float ss = v.x * v.x + v.y * v.y + v.z * v.z + v.w * v.w;
#pragma unroll
    for (int off = 16; off > 0; off >>= 1) ss += __shfl_xor(ss, off, 32);

    const float inv = 1.0f / fmaxf(sqrtf(ss), 1e-12f);
    v.x *= inv; v.y *= inv; v.z *= inv; v.w *= inv;
    v.x = (v.x >= 0.0f) ? v.x : 0.01f * v.x;
    v.y = (v.y >= 0.0f) ? v.y : 0.01f * v.y;
    v.z = (v.z >= 0.0f) ? v.z : 0.01f * v.z;
    v.w = (v.w >= 0.0f) ? v.w : 0.01f * v.w;

    *(float4*)(act + (size_t)node * DHID + k0) = v;
}

// ---------------- final layer: out = accum + sum_r b3[r]  (d == 64) ---------
__global__ void k_bias_out(const float* __restrict__ accum, const float* __restrict__ b,
                           float* __restrict__ out) {
    int i = blockIdx.x * blockDim.x + threadIdx.x;
    if (i >= NN * DOUTF) return;
    int k = i & (DOUTF - 1);
    out[i] = accum[i] + b[k] + b[DOUTF + k] + b[2 * DOUTF + k];
}

// ---------------------------------------------------------------------------
extern "C" void kernel_launch(void* const* d_in, const int* in_sizes, int n_in,
                              void* d_out, int out_size, void* d_ws, size_t ws_size,
                              hipStream_t stream) {
    const float* x   = (const float*)d_in[0];
    const int*   src = (const int*)  d_in[1];
    const int*   dst = (const int*)  d_in[2];
    const float* W1  = (const float*)d_in[3];
    const float* b1  = (const float*)d_in[4];
    const float* W2  = (const float*)d_in[5];
    const float* b2  = (const float*)d_in[6];
    const float* W3  = (const float*)d_in[7];
    const float* b3  = (const float*)d_in[8];
    float* out = (float*)d_out;

    // workspace layout (floats)
    float* f        = (float*)d_ws;
    float* norm_src = f;                          // R*N
    float* norm_dst = norm_src + (size_t)RR * NN; // R*N
    float* Hbuf     = norm_dst + (size_t)RR * NN; // N*128
    float* accum    = Hbuf  + (size_t)NN * DHID;  // N*128
    float* act      = accum + (size_t)NN * DHID;  // N*128

    const int B = 256;

    // --- degree norms (shared by all layers) ---
    {
        int n = 2 * RR * NN;
        k_zero<<<(n + B - 1) / B, B, 0, stream>>>(norm_src, n);
        k_degree<<<(RR * EE + B - 1) / B, B, 0, stream>>>(src, dst, norm_src, norm_dst);
        k_finalize_norm<<<(n + B - 1) / B, B, 0, stream>>>(norm_src, n);
    }

    const int wavesPerBlock = B / 32;
    const int scatterBlocks = (EE + wavesPerBlock - 1) / wavesPerBlock;

    // --- layer 1: x -> act (d=128) ---
    k_zero<<<(NN * DHID + B - 1) / B, B, 0, stream>>>(accum, NN * DHID);
    for (int r = 0; r < RR; ++r) {
        k_gemm_wmma<DHID><<<dim3(NN / 16, DHID / 16), 32, 0, stream>>>(
            x, W1 + (size_t)r * DIN * DHID, norm_src + (size_t)r * NN, Hbuf);
        k_scatter<DHID><<<scatterBlocks, B, 0, stream>>>(
            Hbuf, src + (size_t)r * EE, dst + (size_t)r * EE,
            norm_dst + (size_t)r * NN, accum);
    }
    k_bias_normact<<<(NN + 7) / 8, 256, 0, stream>>>(accum, b1, act);

    // --- layer 2: act -> act (d=128) ---
    k_zero<<<(NN * DHID + B - 1) / B, B, 0, stream>>>(accum, NN * DHID);
    for (int r = 0; r < RR; ++r) {
        k_gemm_wmma<DHID><<<dim3(NN / 16, DHID / 16), 32, 0, stream>>>(
            act, W2 + (size_t)r * DHID * DHID, norm_src + (size_t)r * NN, Hbuf);
        k_scatter<DHID><<<scatterBlocks, B, 0, stream>>>(
            Hbuf, src + (size_t)r * EE, dst + (size_t)r * EE,
            norm_dst + (size_t)r * NN, accum);
    }
    k_bias_normact<<<(NN + 7) / 8, 256, 0, stream>>>(accum, b2, act);

    // --- layer 3: act -> out (d=64) ---
    k_zero<<<(NN * DOUTF + B - 1) / B, B, 0, stream>>>(accum, NN * DOUTF);
    for (int r = 0; r < RR; ++r) {
        k_gemm_wmma<DOUTF><<<dim3(NN / 16, DOUTF / 16), 32, 0, stream>>>(
            act, W3 + (size_t)r * DHID * DOUTF, norm_src + (size_t)r * NN, Hbuf);
        k_scatter<DOUTF><<<scatterBlocks, B, 0, stream>>>(
            Hbuf, src + (size_t)r * EE, dst + (size_t)r * EE,
            norm_dst + (size_t)r * NN, accum);
    }
    k_bias_out<<<(NN * DOUTF + B - 1) / B, B, 0, stream>>>(accum, b3, out);
}